// HeteroGNN_47734266528187
// MI455X (gfx1250) — compile-verified
//
#include <hip/hip_runtime.h>
#include <hip/hip_bf16.h>

typedef __attribute__((ext_vector_type(2))) float v2f;
typedef __attribute__((ext_vector_type(8))) float v8f;

#define H 64
#define TNUM 9
#define LNUM 2

// ---------------------------------------------------------------------------
// Ordered-uint encoding of floats so unsigned atomicMax == float max.
__device__ __forceinline__ unsigned f2o(float f) {
    unsigned u = __float_as_uint(f);
    return (u & 0x80000000u) ? ~u : (u | 0x80000000u);
}
__device__ __forceinline__ float o2f(unsigned u) {
    return (u & 0x80000000u) ? __uint_as_float(u & 0x7fffffffu)
                             : __uint_as_float(~u);
}
// host-side constant: f2o(-inf) = ~0xFF800000 = 0x007FFFFF
#define NEG_INF_ENC 0x007FFFFFu

// ---------------------------------------------------------------------------
// WMMA fp32 GEMM: Out[N,64] (=/+=) X[N,64] @ W[64,64] + bias[64]
// grid.x covers row tiles (2 per block), block = 256 threads = 8 waves
// wave -> (rowTile, colTile): waves 0..3 row tile 0 cols 0..3, waves 4..7 row tile 1
template <bool ACC>
__global__ __launch_bounds__(256) void gemm_n64(const float* __restrict__ X,
                                                const float* __restrict__ W,
                                                const float* __restrict__ bias,
                                                float* __restrict__ Out, int N) {
    int wave = threadIdx.x >> 5;
    int lane = threadIdx.x & 31;
    int rowTile = blockIdx.x * 2 + (wave >> 2);
    int colTile = wave & 3;
    int r0 = rowTile * 16;
    if (r0 >= N) return;  // wave-uniform exit: EXEC stays all-ones for WMMA
    int c0 = colTile * 16;
    int lm = lane & 15;
    bool hi = lane >= 16;

    v8f acc = {};
    const float* xrow = X + (size_t)(r0 + lm) * H;
#pragma unroll
    for (int kk = 0; kk < H; kk += 4) {
        int ka = kk + (hi ? 2 : 0);
        v2f a, b;
        // A 16x4 fp32 layout: lanes 0-15 -> K=kk,kk+1 ; lanes 16-31 -> K=kk+2,kk+3
        a.x = xrow[ka];
        a.y = xrow[ka + 1];
        // B 4x16 fp32: row K striped across lanes (mirrored half-wave split)
        b.x = W[(size_t)ka * H + c0 + lm];
        b.y = W[(size_t)(ka + 1) * H + c0 + lm];
        acc = __builtin_amdgcn_wmma_f32_16x16x4_f32(false, a, false, b,
                                                    (short)0, acc, false, false);
    }
    float bv = bias[c0 + lm];
    int colG = c0 + lm;
    int rbase = r0 + (hi ? 8 : 0);
#pragma unroll
    for (int r = 0; r < 8; ++r) {
        size_t idx = (size_t)(rbase + r) * H + colG;
        float val = acc[r] + bv;
        if (ACC) Out[idx] += val; else Out[idx] = val;
    }
}

// ---------------------------------------------------------------------------
// Pass 1: per-edge raw attention logit + running max per destination.
// One wave (32 lanes) per edge; each lane covers 2 of the 64 channels.
__global__ __launch_bounds__(256) void edge_alpha(const int* __restrict__ src,
                                                  const int* __restrict__ dst,
                                                  const float* __restrict__ ea,
                                                  const float* __restrict__ We,
                                                  const float* __restrict__ q,
                                                  const float* __restrict__ k,
                                                  float* __restrict__ alpha,
                                                  unsigned* __restrict__ mEnc,
                                                  int NE, float scale) {
    int e = blockIdx.x * 8 + (threadIdx.x >> 5);
    if (e >= NE) return;
    int lane = threadIdx.x & 31;
    int s = src[e], d = dst[e];
    float a0 = ea[2 * (size_t)e], a1 = ea[2 * (size_t)e + 1];
    float part = 0.f;
#pragma unroll
    for (int j = 0; j < 2; ++j) {
        int h = lane + 32 * j;
        float eh = a0 * We[h] + a1 * We[H + h];
        float kj = k[(size_t)s * H + h] + eh;
        part += q[(size_t)d * H + h] * kj;
    }
#pragma unroll
    for (int off = 16; off; off >>= 1) part += __shfl_xor(part, off, 32);
    if (lane == 0) {
        float al = part * scale;
        alpha[e] = al;
        atomicMax(mEnc + d, f2o(al));
    }
}

// Pass 2: exponentiate (shifted by segment max), accumulate segment sum.
__global__ __launch_bounds__(256) void edge_exp(const int* __restrict__ dst,
                                                float* __restrict__ alpha,
                                                const unsigned* __restrict__ mEnc,
                                                float* __restrict__ ssum, int NE) {
    int e = blockIdx.x * 256 + threadIdx.x;
    if (e >= NE) return;
    int d = dst[e];
    float ex = __expf(alpha[e] - o2f(mEnc[d]));
    alpha[e] = ex;
    atomicAdd(ssum + d, ex);
}

// Pass 3: weighted scatter of (v[src] + e) into destination accumulator.
__global__ __launch_bounds__(256) void edge_scatter(const int* __restrict__ src,
                                                    const int* __restrict__ dst,
                                                    const float* __restrict__ ea,
                                                    const float* __restrict__ We,
                                                    const float* __restrict__ v,
                                                    const float* __restrict__ alpha,
                                                    const float* __restrict__ ssum,
                                                    float* __restrict__ outs, int NE) {
    int e = blockIdx.x * 8 + (threadIdx.x >> 5);
    if (e >= NE) return;
    int lane = threadIdx.x & 31;
    int s = src[e], d = dst[e];
    float w = alpha[e] / (ssum[d] + 1e-16f);
    float a0 = ea[2 * (size_t)e], a1 = ea[2 * (size_t)e + 1];
#pragma unroll
    for (int j = 0; j < 2; ++j) {
        int h = lane + 32 * j;
        float eh = a0 * We[h] + a1 * We[H + h];
        float vj = v[(size_t)s * H + h] + eh;
        atomicAdd(outs + (size_t)d * H + h, w * vj);
    }
}

// ---------------------------------------------------------------------------
__global__ __launch_bounds__(256) void fill_f32(float* __restrict__ p, float v, size_t n) {
    size_t i = (size_t)blockIdx.x * 256 + threadIdx.x;
    if (i < n) p[i] = v;
}
__global__ __launch_bounds__(256) void fill_u32(unsigned* __restrict__ p, unsigned v, size_t n) {
    size_t i = (size_t)blockIdx.x * 256 + threadIdx.x;
    if (i < n) p[i] = v;
}
__global__ __launch_bounds__(256) void relu_k(float* __restrict__ p, size_t n) {
    size_t i = (size_t)blockIdx.x * 256 + threadIdx.x;
    if (i < n) p[i] = fmaxf(p[i], 0.f);
}

// Final linear on PQ: out[N,4] = x[N,64] @ W[64,4] + b[4]
__global__ __launch_bounds__(256) void final_linear(const float* __restrict__ x,
                                                    const float* __restrict__ W,
                                                    const float* __restrict__ b,
                                                    float* __restrict__ out, int N) {
    int i = blockIdx.x * 256 + threadIdx.x;
    if (i >= N) return;
    float a0 = b[0], a1 = b[1], a2 = b[2], a3 = b[3];
    const float* xr = x + (size_t)i * H;
#pragma unroll
    for (int h = 0; h < H; ++h) {
        float xv = xr[h];
        a0 += xv * W[h * 4 + 0];
        a1 += xv * W[h * 4 + 1];
        a2 += xv * W[h * 4 + 2];
        a3 += xv * W[h * 4 + 3];
    }
    out[(size_t)i * 4 + 0] = a0;
    out[(size_t)i * 4 + 1] = a1;
    out[(size_t)i * 4 + 2] = a2;
    out[(size_t)i * 4 + 3] = a3;
}

// ---------------------------------------------------------------------------
static const int ET_SRC[TNUM] = {0, 0, 0, 1, 1, 1, 2, 2, 3};
static const int ET_DST[TNUM] = {1, 2, 3, 2, 3, 1, 3, 2, 3};

extern "C" void kernel_launch(void* const* d_in, const int* in_sizes, int n_in,
                              void* d_out, int out_size, void* d_ws, size_t ws_size,
                              hipStream_t stream) {
    const float* x_in[4] = {(const float*)d_in[0], (const float*)d_in[1],
                            (const float*)d_in[2], (const float*)d_in[3]};
    const int* edge_index = (const int*)d_in[4];     // [9,2,NE]
    const float* edge_attr = (const float*)d_in[5];  // [9,NE,2]
    const float* Wq = (const float*)d_in[6];
    const float* bq = (const float*)d_in[7];
    const float* Wk = (const float*)d_in[8];
    const float* bk = (const float*)d_in[9];
    const float* Wv = (const float*)d_in[10];
    const float* bv = (const float*)d_in[11];
    const float* We = (const float*)d_in[12];
    const float* Wskip = (const float*)d_in[13];
    const float* bskip = (const float*)d_in[14];
    const float* lin_W = (const float*)d_in[15];
    const float* lin_b = (const float*)d_in[16];

    const int N = in_sizes[0] / H;            // 50000
    const int NE = in_sizes[4] / (TNUM * 2);  // 250000
    const float scale = 1.0f / 8.0f;          // 1/sqrt(64)
    const size_t NH = (size_t)N * H;

    // Workspace carve (floats)
    float* ws = (float*)d_ws;
    size_t off = 0;
    float* bufA[3];
    float* bufB[3];
    for (int i = 0; i < 3; ++i) { bufA[i] = ws + off; off += NH; }  // contiguous
    for (int i = 0; i < 3; ++i) { bufB[i] = ws + off; off += NH; }  // contiguous
    float* qb = ws + off; off += NH;
    float* kb = ws + off; off += NH;
    float* vb = ws + off; off += NH;
    float* alpha = ws + off; off += NE;
    unsigned* mEnc = (unsigned*)(ws + off); off += N;
    float* ssum = ws + off; off += N;
    (void)ws_size;

    // Launch geometry
    const int gemmGrid = (N / 16 + 1) / 2;              // 2 row tiles / block
    const int edgeWGrid = (NE + 7) / 8;                 // wave-per-edge kernels
    const int edgeTGrid = (NE + 255) / 256;             // thread-per-edge
    const int nGrid = (N + 255) / 256;

    const float* xcur[4] = {x_in[0], x_in[1], x_in[2], x_in[3]};

    for (int l = 0; l < LNUM; ++l) {
        float** outs = (l == 0) ? bufA : bufB;
        // zero accumulators for destination types 1..3 (contiguous 3*NH)
        {
            size_t n = 3 * NH;
            fill_f32<<<(int)((n + 255) / 256), 256, 0, stream>>>(outs[0], 0.f, n);
        }
        for (int t = 0; t < TNUM; ++t) {
            const int sT = ET_SRC[t], dT = ET_DST[t];
            const float* xs = xcur[sT];
            const float* xd = xcur[dT];
            const size_t wOff = ((size_t)l * TNUM + t) * H * H;
            const size_t bOff = ((size_t)l * TNUM + t) * H;
            const size_t eOff = ((size_t)l * TNUM + t) * 2 * H;

            gemm_n64<false><<<gemmGrid, 256, 0, stream>>>(xd, Wq + wOff, bq + bOff, qb, N);
            gemm_n64<false><<<gemmGrid, 256, 0, stream>>>(xs, Wk + wOff, bk + bOff, kb, N);
            gemm_n64<false><<<gemmGrid, 256, 0, stream>>>(xs, Wv + wOff, bv + bOff, vb, N);
            gemm_n64<true><<<gemmGrid, 256, 0, stream>>>(xd, Wskip + wOff, bskip + bOff,
                                                         outs[dT - 1], N);

            fill_u32<<<nGrid, 256, 0, stream>>>(mEnc, NEG_INF_ENC, (size_t)N);
            fill_f32<<<nGrid, 256, 0, stream>>>(ssum, 0.f, (size_t)N);

            const int* srcIdx = edge_index + (size_t)t * 2 * NE;
            const int* dstIdx = srcIdx + NE;
            const float* ea = edge_attr + (size_t)t * NE * 2;
            const float* Wet = We + eOff;

            edge_alpha<<<edgeWGrid, 256, 0, stream>>>(srcIdx, dstIdx, ea, Wet, qb, kb,
                                                      alpha, mEnc, NE, scale);
            edge_exp<<<edgeTGrid, 256, 0, stream>>>(dstIdx, alpha, mEnc, ssum, NE);
            edge_scatter<<<edgeWGrid, 256, 0, stream>>>(srcIdx, dstIdx, ea, Wet, vb,
                                                        alpha, ssum, outs[dT - 1], NE);
        }
        // ReLU in place on all three destination-type accumulators (contiguous)
        {
            size_t n = 3 * NH;
            relu_k<<<(int)((n + 255) / 256), 256, 0, stream>>>(outs[0], n);
        }
        // SB (type 0) has no incoming edge types -> passes through unchanged
        xcur[1] = outs[0];
        xcur[2] = outs[1];
        xcur[3] = outs[2];
    }

    final_linear<<<nGrid, 256, 0, stream>>>(xcur[2], lin_W, lin_b, (float*)d_out, N);
}